// CustomHypergraphConv_25718264168628
// MI455X (gfx1250) — compile-verified
//
#include <hip/hip_runtime.h>
#include <hip/hip_bf16.h>

// ---------------------------------------------------------------------------
// Problem constants (match reference)
// ---------------------------------------------------------------------------
constexpr int NUM_NODES = 100000;
constexpr int NUM_EDGES = 25000;
constexpr int NUM_INC   = 800000;
constexpr int DIM       = 128;   // NODE_DIM == EDGE_DIM == HEADS*OUT_CH
constexpr int HEADS     = 4;
constexpr float NEG_SLOPE = 0.2f;

typedef __attribute__((ext_vector_type(2))) float v2f;
typedef __attribute__((ext_vector_type(8))) float v8f;

// ---------------------------------------------------------------------------
// ordered-int encoding so we can do float segment-max with integer atomicMax
// ---------------------------------------------------------------------------
__device__ __forceinline__ int f2o(float f) {
    int i = __float_as_int(f);
    return i >= 0 ? i : (i ^ 0x7fffffff);
}
__device__ __forceinline__ float o2f(int i) {
    int j = i >= 0 ? i : (i ^ 0x7fffffff);
    return __int_as_float(j);
}

// ---------------------------------------------------------------------------
// Init kernels
// ---------------------------------------------------------------------------
__global__ void init_seg_kernel(int* __restrict__ amax, float* __restrict__ denom, int n) {
    int t = blockIdx.x * blockDim.x + threadIdx.x;
    if (t < n) {
        amax[t]  = f2o(-__builtin_inff());
        denom[t] = 0.0f;
    }
}

__global__ void bias_init_kernel(float* __restrict__ out, const float* __restrict__ bias,
                                 long total) {
    long stride = (long)gridDim.x * blockDim.x;
    for (long t = blockIdx.x * (long)blockDim.x + threadIdx.x; t < total; t += stride)
        out[t] = bias[t & (DIM - 1)];
}

// ---------------------------------------------------------------------------
// fp32 WMMA GEMM:  C[M x 128] = A[M x 128] * B[128 x 128]
// Block = 256 threads = 8 waves; block covers 16 rows, wave w covers cols
// [16w, 16w+16).  K marched in steps of 4 via V_WMMA_F32_16X16X4_F32.
// B (the weight matrix, 64 KB) is staged in LDS once per block.
// If C2 != nullptr the epilogue also writes C2 (used to seed edge_aggr).
// ---------------------------------------------------------------------------
__global__ void __launch_bounds__(256)
gemm128_wmma_kernel(const float* __restrict__ A, const float* __restrict__ B,
                    float* __restrict__ C, float* __restrict__ C2, int M) {
    __shared__ float Bs[DIM * DIM];   // 64 KB
    // cooperative load of B into LDS (float4 per thread per step)
    for (int idx = threadIdx.x; idx < DIM * DIM / 4; idx += 256)
        ((float4*)Bs)[idx] = ((const float4*)B)[idx];
    __syncthreads();

    const int wave = threadIdx.x >> 5;
    const int lane = threadIdx.x & 31;
    const int half = lane >> 4;      // 0: K pair {0,1}, 1: K pair {2,3}
    const int lr   = lane & 15;
    const int row0 = blockIdx.x * 16;
    const int col0 = wave * 16;

    const int arow = min(row0 + lr, M - 1);      // clamp; OOB rows never stored
    const float* Arow = A + (long)arow * DIM;

    v8f acc = {};
#pragma unroll
    for (int k0 = 0; k0 < DIM; k0 += 4) {
        const int ka = k0 + 2 * half;
        const float2 av = *(const float2*)(Arow + ka);   // A[m][ka], A[m][ka+1]
        v2f a, b;
        a[0] = av.x;
        a[1] = av.y;
        b[0] = Bs[ka * DIM + col0 + lr];                 // B[ka][n]
        b[1] = Bs[(ka + 1) * DIM + col0 + lr];           // B[ka+1][n]
        acc = __builtin_amdgcn_wmma_f32_16x16x4_f32(
            /*neg_a=*/false, a, /*neg_b=*/false, b,
            /*c_mod=*/(short)0, acc, /*reuse_a=*/false, /*reuse_b=*/false);
    }

#pragma unroll
    for (int r = 0; r < 8; ++r) {
        const int m = row0 + r + 8 * half;   // C/D layout: row = r + 8*(lane/16)
        if (m < M) {
            const float v = acc[r];
            C[(long)m * DIM + col0 + lr] = v;
            if (C2) C2[(long)m * DIM + col0 + lr] = v;
        }
    }
}

// ---------------------------------------------------------------------------
// alpha[row,h] = sum_c feats[row, h*32+c] * attn[h*32+c]
// one wave per row; lane t handles channels [4t,4t+4); head = t>>3;
// reduction across the 8 lanes of each head via width-8 shuffles.
// ---------------------------------------------------------------------------
__global__ void alpha_dot_kernel(const float* __restrict__ feats,
                                 const float* __restrict__ attn,
                                 float* __restrict__ alpha, int nrows) {
    const int gtid = blockIdx.x * blockDim.x + threadIdx.x;
    const int row  = gtid >> 5;
    const int lane = gtid & 31;
    if (row >= nrows) return;
    const float4 f = ((const float4*)(feats + (long)row * DIM))[lane];
    const float4 w = ((const float4*)attn)[lane];
    float d = f.x * w.x + f.y * w.y + f.z * w.z + f.w * w.w;
    d += __shfl_xor(d, 1, 8);
    d += __shfl_xor(d, 2, 8);
    d += __shfl_xor(d, 4, 8);
    if ((lane & 7) == 0) alpha[row * HEADS + (lane >> 3)] = d;
}

// ---------------------------------------------------------------------------
// Softmax pass 1: raw logits + segment max (atomicMax on ordered ints)
// one thread per (incidence, head)
// ---------------------------------------------------------------------------
__global__ void pass1_kernel(const int* __restrict__ node_idx, const int* __restrict__ edge_idx,
                             const float* __restrict__ alpha_l, const float* __restrict__ alpha_r,
                             float* __restrict__ raw, int* __restrict__ amax, int total) {
    const int t = blockIdx.x * blockDim.x + threadIdx.x;
    if (t >= total) return;
    const int i = t >> 2, h = t & 3;
    const int n = node_idx[i], e = edge_idx[i];
    const float v = alpha_l[n * HEADS + h] + alpha_r[e * HEADS + h];
    raw[t] = v;
    atomicMax(&amax[e * HEADS + h], f2o(v));
}

// Softmax pass 2: ex = exp(raw - max); segment sum of ex
__global__ void pass2_kernel(const int* __restrict__ edge_idx, float* __restrict__ raw,
                             const int* __restrict__ amax, float* __restrict__ denom, int total) {
    const int t = blockIdx.x * blockDim.x + threadIdx.x;
    if (t >= total) return;
    const int i = t >> 2, h = t & 3;
    const int e = edge_idx[i];
    const float m  = o2f(amax[e * HEADS + h]);
    const float ex = __expf(raw[t] - m);
    raw[t] = ex;
    atomicAdd(&denom[e * HEADS + h], ex);
}

// ---------------------------------------------------------------------------
// Pass 3: alpha = leaky_relu(ex/denom); scatter node->edge:
//   edge_aggr[e] += node_feats[n] * alpha  (edge_aggr pre-seeded = edge_feats)
// one wave per incidence; lane t handles channels [4t,4t+4), head = t>>3
// ---------------------------------------------------------------------------
__global__ void pass3_kernel(const int* __restrict__ node_idx, const int* __restrict__ edge_idx,
                             const float* __restrict__ node_feats, float* __restrict__ alphabuf,
                             const float* __restrict__ denom, float* __restrict__ edge_aggr,
                             int ninc) {
    const int gtid = blockIdx.x * blockDim.x + threadIdx.x;
    const int i    = gtid >> 5;
    const int lane = gtid & 31;
    if (i >= ninc) return;
    const int h = lane >> 3;
    const int e = edge_idx[i];
    const int n = node_idx[i];
    const float ex = alphabuf[i * HEADS + h];
    const float dn = denom[e * HEADS + h];
    float a = ex / dn;
    a = a > 0.0f ? a : a * NEG_SLOPE;              // leaky_relu AFTER softmax
    if ((lane & 7) == 0) alphabuf[i * HEADS + h] = a;   // keep final alpha for pass 4
    const float4 f = ((const float4*)(node_feats + (long)n * DIM))[lane];
    float* dst = edge_aggr + (long)e * DIM + lane * 4;
    atomicAdd(dst + 0, f.x * a);
    atomicAdd(dst + 1, f.y * a);
    atomicAdd(dst + 2, f.z * a);
    atomicAdd(dst + 3, f.w * a);
}

// Pass 4: scatter edge->node:  out[n] += edge_aggr[e] * alpha  (out pre-seeded = bias)
__global__ void pass4_kernel(const int* __restrict__ node_idx, const int* __restrict__ edge_idx,
                             const float* __restrict__ edge_aggr, const float* __restrict__ alphabuf,
                             float* __restrict__ out, int ninc) {
    const int gtid = blockIdx.x * blockDim.x + threadIdx.x;
    const int i    = gtid >> 5;
    const int lane = gtid & 31;
    if (i >= ninc) return;
    const int h = lane >> 3;
    const int e = edge_idx[i];
    const int n = node_idx[i];
    const float a = alphabuf[i * HEADS + h];
    const float4 f = ((const float4*)(edge_aggr + (long)e * DIM))[lane];
    float* dst = out + (long)n * DIM + lane * 4;
    atomicAdd(dst + 0, f.x * a);
    atomicAdd(dst + 1, f.y * a);
    atomicAdd(dst + 2, f.z * a);
    atomicAdd(dst + 3, f.w * a);
}

// ---------------------------------------------------------------------------
// Host launch
// ---------------------------------------------------------------------------
extern "C" void kernel_launch(void* const* d_in, const int* in_sizes, int n_in,
                              void* d_out, int out_size, void* d_ws, size_t ws_size,
                              hipStream_t stream) {
    (void)in_sizes; (void)n_in; (void)out_size; (void)ws_size;
    const float* x        = (const float*)d_in[0];
    const float* he_attr  = (const float*)d_in[1];
    const int*   node_idx = (const int*)d_in[2];
    const int*   edge_idx = (const int*)d_in[3];
    const float* W_node   = (const float*)d_in[4];
    const float* W_edge   = (const float*)d_in[5];
    const float* attn_l   = (const float*)d_in[6];
    const float* attn_r   = (const float*)d_in[7];
    const float* bias     = (const float*)d_in[8];
    float* out = (float*)d_out;

    // workspace layout (all fp32 unless noted)
    char* ws = (char*)d_ws;
    float* node_feats = (float*)ws;                 ws += sizeof(float) * (size_t)NUM_NODES * DIM;
    float* edge_feats = (float*)ws;                 ws += sizeof(float) * (size_t)NUM_EDGES * DIM;
    float* edge_aggr  = (float*)ws;                 ws += sizeof(float) * (size_t)NUM_EDGES * DIM;
    float* alpha_l    = (float*)ws;                 ws += sizeof(float) * (size_t)NUM_NODES * HEADS;
    float* alpha_r    = (float*)ws;                 ws += sizeof(float) * (size_t)NUM_EDGES * HEADS;
    int*   amax       = (int*)ws;                   ws += sizeof(int)   * (size_t)NUM_EDGES * HEADS;
    float* denom      = (float*)ws;                 ws += sizeof(float) * (size_t)NUM_EDGES * HEADS;
    float* alphabuf   = (float*)ws;                 // NUM_INC * HEADS

    const int B = 256;

    // init segment buffers + output(=bias)
    init_seg_kernel<<<(NUM_EDGES * HEADS + B - 1) / B, B, 0, stream>>>(amax, denom, NUM_EDGES * HEADS);
    bias_init_kernel<<<2048, B, 0, stream>>>(out, bias, (long)NUM_NODES * DIM);

    // projections (WMMA)
    gemm128_wmma_kernel<<<NUM_NODES / 16, B, 0, stream>>>(x, W_node, node_feats, nullptr, NUM_NODES);
    gemm128_wmma_kernel<<<(NUM_EDGES + 15) / 16, B, 0, stream>>>(he_attr, W_edge, edge_feats, edge_aggr, NUM_EDGES);

    // attention logits per row
    alpha_dot_kernel<<<(NUM_NODES * 32 + B - 1) / B, B, 0, stream>>>(node_feats, attn_l, alpha_l, NUM_NODES);
    alpha_dot_kernel<<<(NUM_EDGES * 32 + B - 1) / B, B, 0, stream>>>(edge_feats, attn_r, alpha_r, NUM_EDGES);

    // segment softmax over edges
    const int IH = NUM_INC * HEADS;
    pass1_kernel<<<(IH + B - 1) / B, B, 0, stream>>>(node_idx, edge_idx, alpha_l, alpha_r, alphabuf, amax, IH);
    pass2_kernel<<<(IH + B - 1) / B, B, 0, stream>>>(edge_idx, alphabuf, amax, denom, IH);

    // scatter phases (wave per incidence)
    pass3_kernel<<<(NUM_INC * 32 + B - 1) / B, B, 0, stream>>>(node_idx, edge_idx, node_feats,
                                                               alphabuf, denom, edge_aggr, NUM_INC);
    pass4_kernel<<<(NUM_INC * 32 + B - 1) / B, B, 0, stream>>>(node_idx, edge_idx, edge_aggr,
                                                               alphabuf, out, NUM_INC);
}